// KalmanFilterEstimator_6219112644761
// MI455X (gfx1250) — compile-verified
//
#include <hip/hip_runtime.h>

// Kalman filter scan, persistent single-workgroup kernel for gfx1250.
// All state (x, P, L, temps, u) lives in LDS (~312KB of the 320KB/WGP).
// All GEMMs use V_WMMA_F32_16X16X4_F32 (full fp32 matrix core).
// 128x128 phases: dual-tile register blocking (2 accumulators share B-frags)
// to avoid the 64-VGPR hoisted B-cache that caused scratch spills.

#define NXD 128
#define NYD 64
#define NUD 32
#define NDD 32
#define TSTEPS 2048
#define BBATCH 128
#define NTHREADS 1024

typedef __attribute__((ext_vector_type(2))) float v2f;
typedef __attribute__((ext_vector_type(8))) float v8f;

__device__ __forceinline__ v8f wmma4(v2f a, v2f b, v8f c) {
  // D = A(16x4 f32) * B(4x16 f32) + C(16x16 f32)
  return __builtin_amdgcn_wmma_f32_16x16x4_f32(
      /*neg_a=*/false, a, /*neg_b=*/false, b,
      /*c_mod=*/(short)0, c, /*reuse_a=*/false, /*reuse_b=*/false);
}

__device__ __forceinline__ v2f ld2(const float* p) { return *(const v2f*)p; }

// ---- fragment layouts (ISA 7.12.2, wave32) -------------------------------
// A 16x4 f32 frag: lane l (half=l>>4, lm=l&15): {A[lm][kb+2h], A[lm][kb+2h+1]}
// B 4x16 f32 frag: lane l: {B[kb+2h][lm], B[kb+2h+1][lm]}
// C/D 16x16 f32: VGPR r: lanes0-15 -> M=r, lanes16-31 -> M=r+8; N=lm
// K grouped by 8 (2 wmma per ds_load_2addr_b64-sized fragment batch).

// ---- single-tile variants -------------------------------------------------
__device__ __forceinline__ v8f mm_nn(const float* A, int lda,
                                     const float* Bm, int ldb,
                                     int row, int col, int K, v8f acc, int half) {
  const float* ap = A + row * lda + 2 * half;
#pragma unroll
  for (int kb = 0; kb < K; kb += 8) {
    v2f a0 = ld2(ap + kb);
    v2f a1 = ld2(ap + kb + 4);
    int k0 = kb + 2 * half;
    v2f b0 = {Bm[(k0) * ldb + col], Bm[(k0 + 1) * ldb + col]};
    v2f b1 = {Bm[(k0 + 4) * ldb + col], Bm[(k0 + 5) * ldb + col]};
    acc = wmma4(a0, b0, acc);
    acc = wmma4(a1, b1, acc);
  }
  return acc;
}

__device__ __forceinline__ v8f mm_tn(const float* Am, int ldaT,
                                     const float* Bm, int ldb,
                                     int row, int col, int K, v8f acc, int half) {
#pragma unroll
  for (int kb = 0; kb < K; kb += 8) {
    int k0 = kb + 2 * half;
    v2f a0 = {Am[(k0) * ldaT + row], Am[(k0 + 1) * ldaT + row]};
    v2f a1 = {Am[(k0 + 4) * ldaT + row], Am[(k0 + 5) * ldaT + row]};
    v2f b0 = {Bm[(k0) * ldb + col], Bm[(k0 + 1) * ldb + col]};
    v2f b1 = {Bm[(k0 + 4) * ldb + col], Bm[(k0 + 5) * ldb + col]};
    acc = wmma4(a0, b0, acc);
    acc = wmma4(a1, b1, acc);
  }
  return acc;
}

// ---- dual-tile variants: rows row0 and row1 share every B fragment -------
__device__ __forceinline__ void mm_nn2(const float* A, int lda,
                                       const float* Bm, int ldb,
                                       int row0, int row1, int col, int K,
                                       v8f& acc0, v8f& acc1, int half) {
  const float* ap0 = A + row0 * lda + 2 * half;
  const float* ap1 = A + row1 * lda + 2 * half;
#pragma unroll
  for (int kb = 0; kb < K; kb += 8) {
    v2f a00 = ld2(ap0 + kb);
    v2f a01 = ld2(ap0 + kb + 4);
    v2f a10 = ld2(ap1 + kb);
    v2f a11 = ld2(ap1 + kb + 4);
    int k0 = kb + 2 * half;
    v2f b0 = {Bm[(k0) * ldb + col], Bm[(k0 + 1) * ldb + col]};
    v2f b1 = {Bm[(k0 + 4) * ldb + col], Bm[(k0 + 5) * ldb + col]};
    acc0 = wmma4(a00, b0, acc0);
    acc1 = wmma4(a10, b0, acc1);
    acc0 = wmma4(a01, b1, acc0);
    acc1 = wmma4(a11, b1, acc1);
  }
}

__device__ __forceinline__ void mm_nt2(const float* A, int lda,
                                       const float* Bm, int ldbT,
                                       int row0, int row1, int col, int K,
                                       v8f& acc0, v8f& acc1, int half) {
  const float* ap0 = A + row0 * lda + 2 * half;
  const float* ap1 = A + row1 * lda + 2 * half;
  const float* bp = Bm + col * ldbT + 2 * half;
#pragma unroll
  for (int kb = 0; kb < K; kb += 8) {
    v2f a00 = ld2(ap0 + kb);
    v2f a01 = ld2(ap0 + kb + 4);
    v2f a10 = ld2(ap1 + kb);
    v2f a11 = ld2(ap1 + kb + 4);
    v2f b0 = ld2(bp + kb);
    v2f b1 = ld2(bp + kb + 4);
    acc0 = wmma4(a00, b0, acc0);
    acc1 = wmma4(a10, b0, acc1);
    acc0 = wmma4(a01, b1, acc0);
    acc1 = wmma4(a11, b1, acc1);
  }
}

__device__ __forceinline__ void st_tile(float* D, int ldd, int tm, int tn,
                                        v8f acc, int half, int lm) {
#pragma unroll
  for (int r = 0; r < 8; ++r)
    D[(tm * 16 + 8 * half + r) * ldd + tn * 16 + lm] = acc[r];
}

extern "C" __global__ void __launch_bounds__(NTHREADS, 1)
kalman_scan_kernel(const float* __restrict__ Ym, const float* __restrict__ Mf,
                   const float* __restrict__ DTm, const float* __restrict__ Dd,
                   const float* __restrict__ A_W, const float* __restrict__ B_W,
                   const float* __restrict__ E_W, const float* __restrict__ C_W,
                   const float* __restrict__ Qg, const float* __restrict__ Rg,
                   const float* __restrict__ P0, const float* __restrict__ L0,
                   const float* __restrict__ x0, float* __restrict__ out) {
  extern __shared__ float sm[];
  float* xs   = sm;                    // [128 x 128] state x
  float* Ps   = xs + NXD * NXD;        // [128 x 128] covariance P
  float* tmp  = Ps + NXD * NXD;        // [128 x 128] xp / A@P / C^T@P
  float* Ls   = tmp + NXD * NXD;       // [128 x  64] gain L
  float* bufI = Ls + NXD * NYD;        // [128 x  64] innov / P@C
  float* Saug = bufI + BBATCH * NYD;   // [ 64 x 128] [S | I] Gauss-Jordan
  float* mult = Saug + NYD * 2 * NYD;  // [64]  GJ column multipliers
  float* prow = mult + 64;             // [128] GJ normalized pivot row
  float* us   = prow + 128;            // [128 x 32] u = HEAT_C*mf*dT

  const int tid  = threadIdx.x;
  const int wave = tid >> 5;
  const int lane = tid & 31;
  const int half = lane >> 4;
  const int lm   = lane & 15;
  const float HEAT_C = 0.997f * 4185.5f * (1.0f / 3600.0f);

  // dual-tile mapping for 128x128 phases: (tm2, tn2) and (tm2+4, tn2)
  const int tm2 = wave >> 3;        // 0..3
  const int tn2 = wave & 7;         // 0..7
  const int row0 = tm2 * 16 + lm;   // 0..63
  const int row1 = row0 + 64;       // 64..127
  const int col2 = tn2 * 16 + lm;

  // ---- init carry: x = broadcast(x0), P = P0, L = L0 ----
  for (int i = tid; i < NXD * NXD; i += NTHREADS) {
    xs[i] = x0[i & (NXD - 1)];
    Ps[i] = P0[i];
  }
  for (int i = tid; i < NXD * NYD; i += NTHREADS) Ls[i] = L0[i];
  __syncthreads();

  for (int t = 0; t < TSTEPS; ++t) {
    const float* ym = Ym + (size_t)t * BBATCH * NYD;
    const float* mf = Mf + (size_t)t * BBATCH * NUD;
    const float* dT = DTm + (size_t)t * BBATCH * NUD;
    const float* dd = Dd + (size_t)t * BBATCH * NDD;

    // prefetch next step's streamed inputs into cache (128B granules)
    if (t + 1 < TSTEPS) {
      const float* nym = Ym + (size_t)(t + 1) * BBATCH * NYD;
      const float* nmf = Mf + (size_t)(t + 1) * BBATCH * NUD;
      const float* ndt = DTm + (size_t)(t + 1) * BBATCH * NUD;
      const float* ndd = Dd + (size_t)(t + 1) * BBATCH * NDD;
      if (tid < 256)      __builtin_prefetch(nym + tid * 32, 0, 1);
      else if (tid < 384) __builtin_prefetch(nmf + (tid - 256) * 32, 0, 1);
      else if (tid < 512) __builtin_prefetch(ndt + (tid - 384) * 32, 0, 1);
      else if (tid < 640) __builtin_prefetch(ndd + (tid - 512) * 32, 0, 1);
    }

    // (u) stage u = HEAT_C * mf * dT into LDS (coalesced elementwise pass)
    for (int i = tid; i < BBATCH * NUD; i += NTHREADS)
      us[i] = HEAT_C * mf[i] * dT[i];
    __syncthreads();

    // (a) xp = x@A + u@B + d@E  -> tmp   [128x128], 2 tiles/wave
    {
      v8f acc0 = {}, acc1 = {};
      mm_nn2(xs, NXD, A_W, NXD, row0, row1, col2, NXD, acc0, acc1, half);
      mm_nn2(us, NUD, B_W, NXD, row0, row1, col2, NUD, acc0, acc1, half);
      mm_nn2(dd, NDD, E_W, NXD, row0, row1, col2, NDD, acc0, acc1, half);
      st_tile(tmp, NXD, tm2, tn2, acc0, half, lm);
      st_tile(tmp, NXD, tm2 + 4, tn2, acc1, half, lm);
    }
    __syncthreads();

    // (b) innov = ym - xp@C  -> bufI   [128x64], 1 tile/wave
    {
      int tm = wave >> 2, tn = wave & 3;
      int row = tm * 16 + lm, col = tn * 16 + lm;
      v8f acc = {};
      acc = mm_nn(tmp, NXD, C_W, NYD, row, col, NXD, acc, half);
#pragma unroll
      for (int r = 0; r < 8; ++r) {
        int rr = tm * 16 + 8 * half + r, cc = tn * 16 + lm;
        bufI[rr * NYD + cc] = ym[rr * NYD + cc] - acc[r];
      }
    }
    __syncthreads();

    // (c) x = xp + innov @ L^T   [128x128]  (uses previous-step L)
    {
      v8f acc0 = {}, acc1 = {};
      mm_nt2(bufI, NYD, Ls, NYD, row0, row1, col2, NYD, acc0, acc1, half);
#pragma unroll
      for (int r = 0; r < 8; ++r) {
        int i0 = (tm2 * 16 + 8 * half + r) * NXD + tn2 * 16 + lm;
        int i1 = i0 + 64 * NXD;
        xs[i0] = tmp[i0] + acc0[r];
        xs[i1] = tmp[i1] + acc1[r];
      }
    }
    __syncthreads();

    // (d) AP = A @ P -> tmp   [128x128]
    {
      v8f acc0 = {}, acc1 = {};
      mm_nn2(A_W, NXD, Ps, NXD, row0, row1, col2, NXD, acc0, acc1, half);
      st_tile(tmp, NXD, tm2, tn2, acc0, half, lm);
      st_tile(tmp, NXD, tm2 + 4, tn2, acc1, half, lm);
    }
    __syncthreads();

    // (e) P = AP @ A^T + Q   [128x128]
    {
      v8f acc0 = {}, acc1 = {};
      mm_nt2(tmp, NXD, A_W, NXD, row0, row1, col2, NXD, acc0, acc1, half);
#pragma unroll
      for (int r = 0; r < 8; ++r) {
        int i0 = (tm2 * 16 + 8 * half + r) * NXD + tn2 * 16 + lm;
        int i1 = i0 + 64 * NXD;
        Ps[i0] = acc0[r] + Qg[i0];
        Ps[i1] = acc1[r] + Qg[i1];
      }
    }
    __syncthreads();

    // (f) PC = P @ C -> bufI   [128x64], 1 tile/wave
    {
      int tm = wave >> 2, tn = wave & 3;
      int row = tm * 16 + lm, col = tn * 16 + lm;
      v8f acc = {};
      acc = mm_nn(Ps, NXD, C_W, NYD, row, col, NXD, acc, half);
      st_tile(bufI, NYD, tm, tn, acc, half, lm);
    }
    __syncthreads();

    // (g) Saug = [ R + C^T@PC | I ]   (waves 0-15: S; waves 16-31: identity)
    if (wave < 16) {
      int tm = wave >> 2, tn = wave & 3;
      int row = tm * 16 + lm, col = tn * 16 + lm;
      v8f acc = {};
      acc = mm_tn(C_W, NYD, bufI, NYD, row, col, NXD, acc, half);
#pragma unroll
      for (int r = 0; r < 8; ++r) {
        int rr = tm * 16 + 8 * half + r, cc = tn * 16 + lm;
        Saug[rr * 128 + cc] = acc[r] + Rg[rr * NYD + cc];
      }
    } else {
      int t2 = tid - 512;
      for (int e = t2; e < NYD * NYD; e += 512) {
        int r = e >> 6, c = e & 63;
        Saug[r * 128 + 64 + c] = (r == c) ? 1.0f : 0.0f;
      }
    }
    __syncthreads();

    // (h) Gauss-Jordan: Saug -> [ I | S^-1 ]  (S is SPD; no pivoting)
    //     2 barriers/pivot: read-only pass stages prow/mult, then one write pass.
    for (int p = 0; p < NYD; ++p) {
      float rp = 1.0f / Saug[p * 128 + p];
      if (tid < 128) prow[tid] = Saug[p * 128 + tid] * rp;
      else if (tid < 192) mult[tid - 128] = Saug[(tid - 128) * 128 + p];
      __syncthreads();
      for (int e = tid; e < NYD * 128; e += NTHREADS) {
        int r = e >> 7, c = e & 127;
        Saug[e] = (r == p) ? prow[c] : (Saug[e] - mult[r] * prow[c]);
      }
      __syncthreads();
    }

    // (i) L = PC @ S^-1   [128x64] (Sinv = right half of Saug, ld=128)
    {
      int tm = wave >> 2, tn = wave & 3;
      int row = tm * 16 + lm, col = tn * 16 + lm;
      v8f acc = {};
      acc = mm_nn(bufI, NYD, Saug + 64, 128, row, col, NYD, acc, half);
      st_tile(Ls, NYD, tm, tn, acc, half, lm);
    }
    // (j) CtP = C^T @ P -> tmp  [64x128]  (disjoint buffers from (i))
    {
      int tm = wave >> 3, tn = wave & 7;
      int row = tm * 16 + lm, col = tn * 16 + lm;
      v8f acc = {};
      acc = mm_tn(C_W, NYD, Ps, NXD, row, col, NXD, acc, half);
      st_tile(tmp, NXD, tm, tn, acc, half, lm);
    }
    __syncthreads();

    // (k) P = I - L @ CtP   [128x128]  (as written in the reference)
    {
      v8f acc0 = {}, acc1 = {};
      mm_nn2(Ls, NYD, tmp, NXD, row0, row1, col2, NYD, acc0, acc1, half);
#pragma unroll
      for (int r = 0; r < 8; ++r) {
        int rr0 = tm2 * 16 + 8 * half + r, cc = tn2 * 16 + lm;
        int rr1 = rr0 + 64;
        Ps[rr0 * NXD + cc] = ((rr0 == cc) ? 1.0f : 0.0f) - acc0[r];
        Ps[rr1 * NXD + cc] = ((rr1 == cc) ? 1.0f : 0.0f) - acc1[r];
      }
    }
    __syncthreads();
  }

  // ---- write final x [B x NX] ----
  for (int i = tid; i < BBATCH * NXD; i += NTHREADS) out[i] = xs[i];
}

extern "C" void kernel_launch(void* const* d_in, const int* in_sizes, int n_in,
                              void* d_out, int out_size, void* d_ws, size_t ws_size,
                              hipStream_t stream) {
  (void)in_sizes; (void)n_in; (void)out_size; (void)d_ws; (void)ws_size;
  const float* Ym  = (const float*)d_in[0];
  const float* Mf  = (const float*)d_in[1];
  const float* DTm = (const float*)d_in[2];
  const float* Dd  = (const float*)d_in[3];
  const float* A_W = (const float*)d_in[4];
  const float* B_W = (const float*)d_in[5];
  const float* E_W = (const float*)d_in[6];
  const float* C_W = (const float*)d_in[7];
  const float* Qg  = (const float*)d_in[8];
  const float* Rg  = (const float*)d_in[9];
  const float* P0  = (const float*)d_in[10];
  const float* L0  = (const float*)d_in[11];
  const float* x0  = (const float*)d_in[12];

  const size_t smem =
      (size_t)(3 * NXD * NXD + 2 * NXD * NYD + NYD * 2 * NYD + 64 + 128 +
               BBATCH * NUD) * sizeof(float);
  // ~312 KB dynamic LDS; CDNA5 WGP supports up to 320 KB per workgroup.
  hipFuncSetAttribute((const void*)kalman_scan_kernel,
                      hipFuncAttributeMaxDynamicSharedMemorySize, (int)smem);

  kalman_scan_kernel<<<dim3(1), dim3(NTHREADS), smem, stream>>>(
      Ym, Mf, DTm, Dd, A_W, B_W, E_W, C_W, Qg, Rg, P0, L0, x0, (float*)d_out);
}